// Threshold_1056561955510
// MI455X (gfx1250) — compile-verified
//
#include <hip/hip_runtime.h>

// Elementwise threshold-rounding with reversed output.
//   y[n-1-i] = (x[i] - trunc(x[i]) > th) ? rndne(x[i]) : floor(x[i])
// Memory-bound: 8 B/elem, ~11 us floor at 23.3 TB/s. Strategy: 128-bit
// non-temporal loads/stores, in-register 4-lane reversal so both the load
// and the (descending) store streams stay fully coalesced per wave32.

typedef float v4f __attribute__((ext_vector_type(4)));

__device__ __forceinline__ float thresh_round(float x, float th) {
    float fr = x - __builtin_truncf(x);   // v_trunc_f32 + v_sub_f32 (matches x - trunc(x))
    float r  = __builtin_rintf(x);        // v_rndne_f32: round-half-even == torch.round
    float fl = __builtin_floorf(x);       // v_floor_f32
    return (fr > th) ? r : fl;            // v_cmp_gt_f32 + v_cndmask_b32
}

__global__ void __launch_bounds__(256) threshold_rev_kernel(
        const float* __restrict__ x,
        const float* __restrict__ thr,
        float* __restrict__ out,
        long long n) {
    const float th = thr[0];                         // uniform scalar (L2-cached)
    const long long n4 = n >> 2;
    const long long i4 = (long long)blockIdx.x * blockDim.x + threadIdx.x;

    if (i4 < n4) {
        const v4f* __restrict__ xv = (const v4f*)x;

        // Far-ahead prefetch toward GL2 (gfx1250 global_prefetch_b8 path).
        if (i4 + 4096 < n4) {
            __builtin_prefetch((const void*)&xv[i4 + 4096], 0, 0);
        }

        // 128-bit streaming (non-temporal) load of 4 elements.
        v4f v = __builtin_nontemporal_load(&xv[i4]);

        // Reverse within the quad so the reversed global order is preserved:
        // element x[4*i4+k] lands at out[n-1-(4*i4+k)].
        v4f w;
        w.x = thresh_round(v.w, th);
        w.y = thresh_round(v.z, th);
        w.z = thresh_round(v.y, th);
        w.w = thresh_round(v.x, th);

        // 128-bit streaming store at mirrored quad index; consecutive threads
        // write consecutive (descending) 16B chunks -> coalesced per wave.
        v4f* __restrict__ ov = (v4f*)out;
        __builtin_nontemporal_store(w, &ov[(n4 - 1) - i4]);
    }

    // Generic tail for n % 4 != 0 (empty for n = 2^25, kept for safety).
    const long long base = n4 << 2;
    const long long rem  = n - base;
    if (i4 < rem) {
        float xi = x[base + i4];
        out[n - 1 - (base + i4)] = thresh_round(xi, th);
    }
}

extern "C" void kernel_launch(void* const* d_in, const int* in_sizes, int n_in,
                              void* d_out, int out_size, void* d_ws, size_t ws_size,
                              hipStream_t stream) {
    const float* x   = (const float*)d_in[0];
    const float* thr = (const float*)d_in[1];
    float*       out = (float*)d_out;

    const long long n  = (long long)in_sizes[0];
    const long long n4 = n >> 2;

    const int  block  = 256;                       // 8 wave32s -> fills a WGP
    long long  nthr   = n4 > 0 ? n4 : 1;           // also covers tail threads (rem < 4)
    long long  blocks = (nthr + block - 1) / block;
    if (blocks < 1) blocks = 1;

    threshold_rev_kernel<<<(unsigned)blocks, block, 0, stream>>>(x, thr, out, n);
}